// QuanvolutionHybrid_65481071401719
// MI455X (gfx1250) — compile-verified
//
#include <hip/hip_runtime.h>
#include <math.h>

typedef float v2f __attribute__((ext_vector_type(2)));
typedef float v8f __attribute__((ext_vector_type(8)));

// fc_w staged in LDS as 16 rows (classes, rows 10..15 zero) x 784 cols (k),
// padded to stride 788 dwords: gcd(788,64)=4 so the 16 lanes of each half-wave
// land on 16 distinct banks, and the two halves interleave -> conflict-free b64.
#define WLDS_STRIDE 788

__global__ __launch_bounds__(256)
void quanv_fused_kernel(const float* __restrict__ x,
                        const float* __restrict__ vqc,
                        const float* __restrict__ fc_w,
                        const float* __restrict__ fc_b,
                        const float* __restrict__ bias,
                        float* __restrict__ out,
                        int nTiles)
{
    __shared__ float wlds[16 * WLDS_STRIDE];

    const int tid = threadIdx.x;

    // Stage fc_w (10x784) into padded 16x788 LDS image; rows 10..15 = 0.
    for (int i = tid; i < 16 * 784; i += 256) {
        const int row = i / 784;
        const int col = i - row * 784;
        wlds[row * WLDS_STRIDE + col] = (row < 10) ? fc_w[row * 784 + col] : 0.0f;
    }
    __syncthreads();

    const int lane = tid & 31;
    const int wave = tid >> 5;
    const int tile = blockIdx.x * 8 + wave;   // 16 batch rows per wave
    if (tile >= nTiles) return;               // wave-uniform exit

    const int m = lane & 15;  // A: batch row within tile; B/D: class column N
    const int h = lane >> 4;  // 0 -> wires (0,1) [K=0,1], 1 -> wires (2,3) [K=2,3]

    // Circuit params for this half-wave's wire pair.
    const float pa  = vqc[2 * h + 0];
    const float pb  = vqc[2 * h + 1];
    const float cpa = __cosf(pa), spa = __sinf(pa);
    const float cpb = __cosf(pb), spb = __sinf(pb);

    const int img = tile * 16 + m;
    const float* __restrict__ ximg = x + (size_t)img * 784;
    const float* __restrict__ wrow = &wlds[m * WLDS_STRIDE + 2 * h];

    v8f acc = {0.f, 0.f, 0.f, 0.f, 0.f, 0.f, 0.f, 0.f};

    int p = 0;
    for (int pr = 0; pr < 14; ++pr) {
        // half 0 reads image row 2*pr, half 1 reads row 2*pr+1
        const float* __restrict__ prow = ximg + (2 * pr + h) * 28;
        #pragma unroll
        for (int pc = 0; pc < 14; ++pc, ++p) {
            // --- A fragment: (z_even, z_odd) for this half's wire pair ---
            const v2f xv = *(const v2f*)(prow + 2 * pc);          // b64, 8B aligned
            const float sa = __sinf(xv.x), ca = __cosf(xv.x);
            const float sb = __sinf(xv.y), cb = __cosf(xv.y);
            v2f a;
            a.x = cpa * ca + spa * sa * sb;        // z0 (h=0) / z2 (h=1): K = 2h
            a.y = cpb * ca * cb + spb * sb;        // z1 (h=0) / z3 (h=1): K = 2h+1
            // --- B fragment: fc_w[class=m][k = 4p + 2h + {0,1}] from LDS ---
            const v2f b = *(const v2f*)(wrow + 4 * p);            // ds_load_b64
            // D(16x16) += A(16x4) x B(4x16)
            acc = __builtin_amdgcn_wmma_f32_16x16x4_f32(
                      /*neg_a=*/false, a, /*neg_b=*/false, b,
                      /*c_mod=*/(short)0, acc,
                      /*reuse_a=*/false, /*reuse_b=*/false);
        }
    }

    // Epilogue: logits = acc + fc_b - bias, then row-wise log_softmax.
    // D layout: VGPR v, lanes 0-15 -> M=v, lanes 16-31 -> M=v+8; N = lane%16.
    const bool  valid = (m < 10);
    const float badd  = valid ? (fc_b[m] - bias[m]) : 0.0f;

    #pragma unroll
    for (int v = 0; v < 8; ++v) {
        const int   grow  = tile * 16 + v + 8 * h;   // global batch row
        const float logit = acc[v] + badd;

        float mx = valid ? logit : -__builtin_huge_valf();
        #pragma unroll
        for (int s = 8; s >= 1; s >>= 1)
            mx = fmaxf(mx, __shfl_xor(mx, s, 16));   // reduce over N within half

        float e = valid ? __expf(logit - mx) : 0.0f;
        float sum = e;
        #pragma unroll
        for (int s = 8; s >= 1; s >>= 1)
            sum += __shfl_xor(sum, s, 16);

        if (valid)
            out[(size_t)grow * 10 + m] = logit - mx - __logf(sum);
    }
}

extern "C" void kernel_launch(void* const* d_in, const int* in_sizes, int n_in,
                              void* d_out, int out_size, void* d_ws, size_t ws_size,
                              hipStream_t stream) {
    const float* x    = (const float*)d_in[0];  // [B,1,28,28]
    const float* vqc  = (const float*)d_in[1];  // [4]
    const float* fc_w = (const float*)d_in[2];  // [10,784]
    const float* fc_b = (const float*)d_in[3];  // [10]
    const float* bias = (const float*)d_in[4];  // [10]
    float* out = (float*)d_out;                 // [B,10] log-softmax

    const int B      = in_sizes[0] / 784;
    const int nTiles = B / 16;                  // 16 batch rows per wave
    const int blocks = (nTiles + 7) / 8;        // 8 waves (tiles) per 256-thr block

    quanv_fused_kernel<<<blocks, 256, 0, stream>>>(x, vqc, fc_w, fc_b, bias, out, nTiles);
}